// GraphUNet_no_TopK_19524921328239
// MI455X (gfx1250) — compile-verified
//
#include <hip/hip_runtime.h>
#include <hip/hip_bf16.h>

typedef float v2f __attribute__((ext_vector_type(2)));
typedef float v8f __attribute__((ext_vector_type(8)));

// ---------------- elementwise / graph kernels ----------------

__global__ void k_zero(float* p, int n) {
    int i = blockIdx.x * blockDim.x + threadIdx.x;
    if (i < n) p[i] = 0.f;
}

__global__ void k_copy(float* dst, const float* src, int n) {
    int i = blockIdx.x * blockDim.x + threadIdx.x;
    if (i < n) dst[i] = src[i];
}

// deg[row] += 1 per edge (deg buffer pre-zeroed)
__global__ void k_deg(float* deg, const int* ei, int E) {
    int e = blockIdx.x * blockDim.x + threadIdx.x;
    if (e < E) atomicAdd(&deg[ei[e]], 1.0f);
}

// deg -> d^{-1/2}
__global__ void k_dinv(float* dinv, int n) {
    int i = blockIdx.x * blockDim.x + threadIdx.x;
    if (i < n) {
        float d = dinv[i];
        dinv[i] = (d > 0.f) ? rsqrtf(d) : 0.f;
    }
}

// ew[e] = -(dinv[row]*dinv[col]);  ei = [row(E) | col(E)]
__global__ void k_ew(float* ew, const float* dinv, const int* ei, int E) {
    int e = blockIdx.x * blockDim.x + threadIdx.x;
    if (e < E) ew[e] = -(dinv[ei[e]] * dinv[ei[E + e]]);
}

// A[:,0:C] = src (contiguous [n,C]); A pre-zeroed (incl. row padding)
__global__ void k_fill_contig(float* A, const float* src, int n, int C) {
    int tid = blockIdx.x * blockDim.x + threadIdx.x;
    if (tid < n * C) {
        int i = tid / C, c = tid % C;
        A[i * 3 * C + c] = src[tid];
    }
}

// A[:,0:Cup] = hex_up(h,up); A[:,Cup:Cup+Csk] = skip
__global__ void k_fill_upconcat(float* A, const float* h, const int* up,
                                const float* skip, int n, int n_prev,
                                int Cup, int Csk) {
    int Cin = Cup + Csk;
    int tid = blockIdx.x * blockDim.x + threadIdx.x;
    if (tid >= n * Cin) return;
    int i = tid / Cin, c = tid % Cin;
    float v;
    if (c < Cup) {
        if (i < n_prev) {
            v = h[i * Cup + c];
        } else {
            int j = i - n_prev;
            int a0 = up[2 * j], a1 = up[2 * j + 1];
            v = 0.5f * (h[a0 * Cup + c] + h[a1 * Cup + c]);
        }
    } else {
        v = skip[i * Csk + (c - Cup)];
    }
    A[i * 3 * Cin + c] = v;
}

// A[row, dstOff+c] += ew[e] * A[col, srcOff+c]   (stride 3C)
__global__ void k_prop(float* A, const float* ew, const int* ei,
                       int E, int C, int srcOff, int dstOff) {
    int tid = blockIdx.x * blockDim.x + threadIdx.x;
    if (tid >= E * C) return;
    int e = tid / C, c = tid % C;
    int r = ei[e], col = ei[E + e];
    atomicAdd(&A[r * 3 * C + dstOff + c], ew[e] * A[col * 3 * C + srcOff + c]);
}

// t2 = 2*acc - x
__global__ void k_t2fin(float* A, int n, int C) {
    int tid = blockIdx.x * blockDim.x + threadIdx.x;
    if (tid >= n * C) return;
    int i = tid / C, c = tid % C;
    int b = i * 3 * C;
    A[b + 2 * C + c] = 2.f * A[b + 2 * C + c] - A[b + c];
}

// stage W[K,Cout] -> Wb in B-fragment order, zero-padded to Coutp columns:
//   Wb[(kb*Coutp + col)*4 + kl] = W[(4*kb+kl)*Cout + col]
__global__ void k_stagew(float* Wb, const float* W, int K, int Cout, int Coutp) {
    int tid = blockIdx.x * blockDim.x + threadIdx.x;
    int tot = (K >> 2) * Coutp * 4;
    if (tid >= tot) return;
    int kb  = tid / (Coutp * 4);
    int rem = tid - kb * Coutp * 4;
    int col = rem >> 2;
    int kl  = rem & 3;
    Wb[tid] = (col < Cout) ? W[(4 * kb + kl) * Cout + col] : 0.f;
}

// ---------------- WMMA GEMM: H = A(n x K) * W(K x Cout) + bias, opt ReLU ----
// One wave computes a 64x16 output strip (4 accumulators) with
// V_WMMA_F32_16X16X4_F32. A is zero-padded to 64-row multiples, Wb is the
// staged fragment-ordered weight panel. Fragment loads are double-buffered
// one K-step ahead so the WMMAs only wait on loads issued a full iteration
// earlier (partial s_wait_loadcnt instead of a drain).
__global__ void k_cheb_gemm(const float* __restrict__ A, const float* __restrict__ Wb,
                            const float* __restrict__ bias, float* __restrict__ H,
                            int n, int K, int Cout, int Coutp, int do_relu) {
    const int lane = threadIdx.x & 31;
    const int half = lane >> 4;          // 0|1 -> K offset 0|2 (ISA f32 16x4 layout)
    const int l16  = lane & 15;
    const int m0 = blockIdx.x * 64;
    const int n0 = blockIdx.y * 16;
    const int bcol = n0 + l16;

    const float* Ab = A + (m0 + l16) * K + 2 * half;
    const float* Bb = Wb + (n0 + l16) * 4 + 2 * half;
    const int bstep = Coutp * 4;
    const int Kb = K >> 2;               // always >= 3 here (K >= 12)

    v8f acc[4];
#pragma unroll
    for (int t = 0; t < 4; ++t) acc[t] = v8f{0.f,0.f,0.f,0.f,0.f,0.f,0.f,0.f};

    // prologue: load step 0
    v2f cb  = *(const v2f*)(Bb);
    v2f ca0 = *(const v2f*)(Ab);
    v2f ca1 = *(const v2f*)(Ab + 16 * K);
    v2f ca2 = *(const v2f*)(Ab + 32 * K);
    v2f ca3 = *(const v2f*)(Ab + 48 * K);

#pragma unroll 1
    for (int kb = 0; kb < Kb - 1; ++kb) {
        Bb += bstep;
        Ab += 4;
        // issue next step's loads first ...
        v2f nb  = *(const v2f*)(Bb);
        v2f na0 = *(const v2f*)(Ab);
        v2f na1 = *(const v2f*)(Ab + 16 * K);
        v2f na2 = *(const v2f*)(Ab + 32 * K);
        v2f na3 = *(const v2f*)(Ab + 48 * K);
        // ... then consume current step (loads from previous iteration)
        acc[0] = __builtin_amdgcn_wmma_f32_16x16x4_f32(false, ca0, false, cb, (short)0, acc[0], false, false);
        acc[1] = __builtin_amdgcn_wmma_f32_16x16x4_f32(false, ca1, false, cb, (short)0, acc[1], false, false);
        acc[2] = __builtin_amdgcn_wmma_f32_16x16x4_f32(false, ca2, false, cb, (short)0, acc[2], false, false);
        acc[3] = __builtin_amdgcn_wmma_f32_16x16x4_f32(false, ca3, false, cb, (short)0, acc[3], false, false);
        cb = nb; ca0 = na0; ca1 = na1; ca2 = na2; ca3 = na3;
    }
    // epilogue step
    acc[0] = __builtin_amdgcn_wmma_f32_16x16x4_f32(false, ca0, false, cb, (short)0, acc[0], false, false);
    acc[1] = __builtin_amdgcn_wmma_f32_16x16x4_f32(false, ca1, false, cb, (short)0, acc[1], false, false);
    acc[2] = __builtin_amdgcn_wmma_f32_16x16x4_f32(false, ca2, false, cb, (short)0, acc[2], false, false);
    acc[3] = __builtin_amdgcn_wmma_f32_16x16x4_f32(false, ca3, false, cb, (short)0, acc[3], false, false);

    const bool colOK = bcol < Cout;
#pragma unroll
    for (int t = 0; t < 4; ++t) {
#pragma unroll
        for (int r = 0; r < 8; ++r) {
            int orow = m0 + 16 * t + r + 8 * half;   // C/D layout: VGPR r -> M=r(+8)
            if (orow < n && colOK) {
                float v = acc[t][r] + bias[bcol];
                if (do_relu) v = fmaxf(v, 0.f);
                H[orow * Cout + bcol] = v;
            }
        }
    }
}

// ---------------- softmax over 37 classes, one thread per row --------------
__global__ void k_softmax(const float* H, float* out, int n, int C) {
    int i = blockIdx.x * blockDim.x + threadIdx.x;
    if (i >= n) return;
    const float* r = H + i * C;
    float m = r[0];
    for (int c = 1; c < C; ++c) m = fmaxf(m, r[c]);
    float s = 0.f;
    for (int c = 0; c < C; ++c) s += __expf(r[c] - m);
    float inv = 1.f / s;
    float* o = out + i * C;
    for (int c = 0; c < C; ++c) o[c] = __expf(r[c] - m) * inv;
}

// ---------------- host-side layer driver ----------------

static inline int cdiv(int a, int b) { return (a + b - 1) / b; }

static void run_conv(hipStream_t s, float* A, float* Wb, float* dinv, float* ew,
                     const int* ei, int E, int n, int C, int Cout,
                     const float* W, const float* bias, float* H, bool relu,
                     const float* src_contig,                     // mode 1
                     const float* h_prev, const int* up,          // mode 2
                     const float* skip, int n_prev, int Cup, int Csk) {
    const int T = 256;
    const int K = 3 * C;
    const int n64 = cdiv(n, 64) * 64;
    const int Coutp = cdiv(Cout, 16) * 16;

    k_zero <<<cdiv(n, T), T, 0, s>>>(dinv, n);
    k_deg  <<<cdiv(E, T), T, 0, s>>>(dinv, ei, E);
    k_dinv <<<cdiv(n, T), T, 0, s>>>(dinv, n);
    k_ew   <<<cdiv(E, T), T, 0, s>>>(ew, dinv, ei, E);

    k_zero <<<cdiv(n64 * K, T), T, 0, s>>>(A, n64 * K);   // incl. row padding
    if (src_contig)
        k_fill_contig<<<cdiv(n * C, T), T, 0, s>>>(A, src_contig, n, C);
    else
        k_fill_upconcat<<<cdiv(n * C, T), T, 0, s>>>(A, h_prev, up, skip, n,
                                                     n_prev, Cup, Csk);
    k_prop <<<cdiv(E * C, T), T, 0, s>>>(A, ew, ei, E, C, 0, C);       // t1
    k_prop <<<cdiv(E * C, T), T, 0, s>>>(A, ew, ei, E, C, C, 2 * C);   // acc for t2
    k_t2fin<<<cdiv(n * C, T), T, 0, s>>>(A, n, C);

    k_stagew<<<cdiv(K * Coutp, T), T, 0, s>>>(Wb, W, K, Cout, Coutp);
    dim3 g(cdiv(n, 64), Coutp / 16);
    k_cheb_gemm<<<g, 32, 0, s>>>(A, Wb, bias, H, n, K, Cout, Coutp, relu ? 1 : 0);
}

extern "C" void kernel_launch(void* const* d_in, const int* in_sizes, int n_in,
                              void* d_out, int out_size, void* d_ws, size_t ws_size,
                              hipStream_t stream) {
    (void)in_sizes; (void)n_in; (void)out_size; (void)ws_size;
    const float* x = (const float*)d_in[0];
    const int* e[7];  for (int l = 1; l <= 6; ++l) e[l]  = (const int*)d_in[l];
    const int* up[7]; for (int l = 2; l <= 6; ++l) up[l] = (const int*)d_in[5 + l];
    const float *w[12], *b[12];
    for (int i = 1; i <= 11; ++i) {
        w[i] = (const float*)d_in[10 + 2 * i];
        b[i] = (const float*)d_in[11 + 2 * i];
    }
    const int E1 = 240, E2 = 960, E3 = 3840, E4 = 15360, E5 = 61440, E6 = 245760;

    // workspace layout (floats)
    float* ws   = (float*)d_ws;
    float* dinv = ws;                          // 40962
    float* ew   = dinv + 40962;                // 245760
    float* A    = ew + 245760;                 // 41024*108 max (64-row padded)
    float* Wb   = A + 41024 * 108;             // 1536*512 max staged weights
    float* Ha   = Wb + 1536 * 512;             // 40962*37 max
    float* Hb   = Ha + 40962 * 37;
    float* s1   = Hb + 40962 * 37;             // 10242*32
    float* s2   = s1 + 10242 * 32;             // 2562*64
    float* s3   = s2 + 2562 * 64;              // 642*128
    float* s4   = s3 + 642 * 128;              // 162*256

    const int T = 256;

    // ---- encoder ----
    run_conv(stream, A, Wb, dinv, ew, e[6], E6, 40962, 4, 32, w[1], b[1], Ha, true,
             x, nullptr, nullptr, nullptr, 0, 0, 0);
    k_copy<<<cdiv(10242 * 32, T), T, 0, stream>>>(s1, Ha, 10242 * 32);   // x1
    run_conv(stream, A, Wb, dinv, ew, e[5], E5, 10242, 32, 64, w[2], b[2], Hb, true,
             Ha, nullptr, nullptr, nullptr, 0, 0, 0);
    k_copy<<<cdiv(2562 * 64, T), T, 0, stream>>>(s2, Hb, 2562 * 64);     // x2
    run_conv(stream, A, Wb, dinv, ew, e[4], E4, 2562, 64, 128, w[3], b[3], Ha, true,
             Hb, nullptr, nullptr, nullptr, 0, 0, 0);
    k_copy<<<cdiv(642 * 128, T), T, 0, stream>>>(s3, Ha, 642 * 128);     // x3
    run_conv(stream, A, Wb, dinv, ew, e[3], E3, 642, 128, 256, w[4], b[4], Hb, true,
             Ha, nullptr, nullptr, nullptr, 0, 0, 0);
    k_copy<<<cdiv(162 * 256, T), T, 0, stream>>>(s4, Hb, 162 * 256);     // x4
    run_conv(stream, A, Wb, dinv, ew, e[2], E2, 162, 256, 512, w[5], b[5], Ha, true,
             Hb, nullptr, nullptr, nullptr, 0, 0, 0);
    // x5 = Ha[:42] (contiguous prefix)
    run_conv(stream, A, Wb, dinv, ew, e[1], E1, 42, 512, 512, w[6], b[6], Hb, true,
             Ha, nullptr, nullptr, nullptr, 0, 0, 0);

    // ---- decoder (hex_up + skip concat) ----
    run_conv(stream, A, Wb, dinv, ew, e[2], E2, 162, 768, 256, w[7], b[7], Ha, true,
             nullptr, Hb, up[2], s4, 42, 512, 256);
    run_conv(stream, A, Wb, dinv, ew, e[3], E3, 642, 384, 128, w[8], b[8], Hb, true,
             nullptr, Ha, up[3], s3, 162, 256, 128);
    run_conv(stream, A, Wb, dinv, ew, e[4], E4, 2562, 192, 64, w[9], b[9], Ha, true,
             nullptr, Hb, up[4], s2, 642, 128, 64);
    run_conv(stream, A, Wb, dinv, ew, e[5], E5, 10242, 96, 32, w[10], b[10], Hb, true,
             nullptr, Ha, up[5], s1, 2562, 64, 32);
    run_conv(stream, A, Wb, dinv, ew, e[6], E6, 40962, 36, 37, w[11], b[11], Ha, false,
             nullptr, Hb, up[6], x, 10242, 32, 4);

    k_softmax<<<cdiv(40962, T), T, 0, stream>>>(Ha, (float*)d_out, 40962, 37);
}